// CalAttn_7327214207255
// MI455X (gfx1250) — compile-verified
//
#include <hip/hip_runtime.h>
#include <math.h>

// ---------------------------------------------------------------------------
// Attention with full attn-matrix output, fp32, using V_WMMA_F32_16X16X4_F32.
// B=4 H=12 S=2048 D=64.  Grid: x = S/16 q-tiles (128), y = B*H (48).
// Block: 128 threads = 4 waves (wave32).
// attn/out stores are non-temporal: 805MB streaming write must not evict the
// reused K/V/mask working set from the 192MB L2.
// ---------------------------------------------------------------------------

typedef __attribute__((ext_vector_type(2))) float v2f;
typedef __attribute__((ext_vector_type(8))) float v8f;

#define S_LEN      2048
#define D_K        64
#define HEADS      12
#define QT         16          // query rows per workgroup
#define NWAVE      4
#define NTHREADS   (NWAVE * 32)
#define LDS_STRIDE 2052        // 2048 + 4 pad floats (banks shift by 4/row)
#define NEG_VAL    (-1000000000.0f)

// dynamic LDS partition (floats):
//   sc    : QT * LDS_STRIDE      (score / prob tile)
//   red   : 16 * 8               (row reduction scratch)
//   rowinv: 16
#define SMEM_FLOATS (QT * LDS_STRIDE + 16 * 8 + 16)

__global__ void __launch_bounds__(NTHREADS)
attn_fp32_wmma_kernel(const float* __restrict__ q,
                      const float* __restrict__ k,
                      const float* __restrict__ v,
                      const unsigned char* __restrict__ mask,
                      float* __restrict__ out,
                      float* __restrict__ attn)
{
    extern __shared__ float smem[];
    float* sc     = smem;                       // [QT][LDS_STRIDE]
    float* red    = sc + QT * LDS_STRIDE;       // [16][8]
    float* rowinv = red + 16 * 8;               // [16]

    const int tid   = threadIdx.x;
    const int wave  = tid >> 5;
    const int lane  = tid & 31;
    const int bh    = blockIdx.y;               // 0..47
    const int b     = bh / HEADS;
    const int qbase = blockIdx.x * QT;

    const int lrow  = lane & 15;                // M (A / score-N) row selector
    const int koff  = (lane >> 4) * 2;          // K sub-pair per half-wave
    const int mhi   = (lane >> 4) << 3;         // +8 rows for upper half-wave

    // ---------------- Phase 0: preload Q tile A-operands -------------------
    // A layout (32-bit 16x4): lanes 0-15 -> M=lane, K = k0+{0,1}
    //                         lanes16-31 -> M=lane-16, K = k0+{2,3}
    const v2f* qrow = (const v2f*)(q + ((size_t)(bh * S_LEN + qbase + lrow)) * D_K + koff);
    v2f qa[16];
#pragma unroll
    for (int i = 0; i < 16; ++i)
        qa[i] = qrow[2 * i];                    // floats at 4*i+koff, +1

    // ---------------- Phase 1: scores = QK^T / sqrt(D), mask, -> LDS -------
    for (int kt = wave; kt < S_LEN / 16; kt += NWAVE) {
        const int kbase = kt * 16;
        // B layout (4x16): VGPR0 lanes0-15 = row K=k0+0, lanes16-31 = k0+2;
        //                  VGPR1 lanes0-15 = k0+1,       lanes16-31 = k0+3.
        // B[kk][n] = K[kbase+n][kk]  -> lane loads 2 consecutive dims of key row n.
        const v2f* krow = (const v2f*)(k + ((size_t)(bh * S_LEN + kbase + lrow)) * D_K + koff);

        v8f acc = {0.f, 0.f, 0.f, 0.f, 0.f, 0.f, 0.f, 0.f};
#pragma unroll
        for (int i = 0; i < 16; ++i) {
            v2f bb = krow[2 * i];
            acc = __builtin_amdgcn_wmma_f32_16x16x4_f32(
                false, qa[i], false, bb, (short)0, acc, false, false);
        }

        const int col = kbase + lrow;
        const unsigned char* mrow = mask + ((size_t)(b * S_LEN + qbase + mhi)) * S_LEN + col;
#pragma unroll
        for (int r = 0; r < 8; ++r) {
            float sv = acc[r] * 0.125f;                     // 1/sqrt(64)
            if (mrow[(size_t)r * S_LEN]) sv = NEG_VAL;      // mask rows m = r + mhi
            sc[(r + mhi) * LDS_STRIDE + col] = sv;
        }
    }
    __syncthreads();

    // ---------------- Phase 2: row softmax over 2048 cols -------------------
    {
        const int r = tid & 15;                 // row
        const int j = tid >> 4;                 // 8 threads per row
        float* row = sc + r * LDS_STRIDE;

        float mx = -INFINITY;
        for (int c = j; c < S_LEN; c += 8)
            mx = fmaxf(mx, row[c]);
        red[r * 8 + j] = mx;
        __syncthreads();

        float rmax = red[r * 8 + 0];
#pragma unroll
        for (int jj = 1; jj < 8; ++jj)
            rmax = fmaxf(rmax, red[r * 8 + jj]);
        __syncthreads();

        float psum = 0.f;
        for (int c = j; c < S_LEN; c += 8) {
            float e = expf(row[c] - rmax);
            row[c] = e;
            psum += e;
        }
        red[r * 8 + j] = psum;
        __syncthreads();

        if (j == 0) {
            float s = 0.f;
#pragma unroll
            for (int jj = 0; jj < 8; ++jj)
                s += red[r * 8 + jj];
            rowinv[r] = 1.0f / s;
        }
        __syncthreads();
    }

    // normalize + coalesced non-temporal global store of attn tile
    {
        float* attn_tile = attn + ((size_t)(bh * S_LEN + qbase)) * S_LEN;
        for (int idx = tid; idx < QT * S_LEN; idx += NTHREADS) {
            const int r2 = idx >> 11;           // / 2048
            const int c2 = idx & (S_LEN - 1);
            float p = sc[r2 * LDS_STRIDE + c2] * rowinv[r2];
            sc[r2 * LDS_STRIDE + c2] = p;
            __builtin_nontemporal_store(p, &attn_tile[(size_t)r2 * S_LEN + c2]);
        }
        __syncthreads();
    }

    // ---------------- Phase 3: out = P @ V  (wave owns 16 cols of D) -------
    {
        const int dbase = wave * 16;
        const float* vcol = v + (size_t)bh * S_LEN * D_K + dbase + lrow;

        v8f acc = {0.f, 0.f, 0.f, 0.f, 0.f, 0.f, 0.f, 0.f};
        for (int k0 = 0; k0 < S_LEN; k0 += 4) {
            // A from LDS probs: lane half-wave picks K = k0+koff,+1  (8B aligned)
            v2f a = *(const v2f*)&sc[lrow * LDS_STRIDE + k0 + koff];
            // B = V[k0+koff + {0,1}][dbase + lrow]
            float b0 = vcol[(size_t)(k0 + koff + 0) * D_K];
            float b1 = vcol[(size_t)(k0 + koff + 1) * D_K];
            v2f bb = {b0, b1};
            acc = __builtin_amdgcn_wmma_f32_16x16x4_f32(
                false, a, false, bb, (short)0, acc, false, false);
        }

        float* orow = out + ((size_t)(bh * S_LEN + qbase + mhi)) * D_K + dbase + lrow;
#pragma unroll
        for (int r = 0; r < 8; ++r)
            __builtin_nontemporal_store(acc[r], &orow[(size_t)r * D_K]);
    }
}

extern "C" void kernel_launch(void* const* d_in, const int* in_sizes, int n_in,
                              void* d_out, int out_size, void* d_ws, size_t ws_size,
                              hipStream_t stream) {
    (void)in_sizes; (void)n_in; (void)d_ws; (void)ws_size; (void)out_size;

    const float* q = (const float*)d_in[0];
    const float* k = (const float*)d_in[1];
    const float* v = (const float*)d_in[2];
    const unsigned char* mask = (const unsigned char*)d_in[3];  // jnp.bool_ = 1 byte

    const int B = 4;
    float* out  = (float*)d_out;                                // (B,H,S,D) first
    float* attn = out + (size_t)B * HEADS * S_LEN * D_K;        // then (B,H,S,S)

    dim3 grid(S_LEN / QT, B * HEADS);      // (128, 48)
    dim3 block(NTHREADS);                  // 128 threads = 4 wave32
    size_t smem = SMEM_FLOATS * sizeof(float);   // ~132 KB dynamic LDS

    attn_fp32_wmma_kernel<<<grid, block, smem, stream>>>(q, k, v, mask, out, attn);
}